// AdditiveAttention_35854386987769
// MI455X (gfx1250) — compile-verified
//
#include <hip/hip_runtime.h>
#include <hip/hip_bf16.h>

// Problem constants (from reference): B=16, Q=128, K=512, H=256, D=256
#define BB   16
#define QQ   128
#define KK   512
#define HH   256
#define DD   256
#define NEGV (-1e6f)

typedef __attribute__((ext_vector_type(2))) float v2f;
typedef __attribute__((ext_vector_type(8))) float v8f;

// CDNA5 f32 WMMA: D(16x16 f32) = A(16x4 f32) x B(4x16 f32) + C
#define WMMA_F32_4(a, b, c) \
  __builtin_amdgcn_wmma_f32_16x16x4_f32(false, (a), false, (b), (short)0, (c), false, false)

__device__ __forceinline__ float fast_tanh(float x) {
#if __has_builtin(__builtin_amdgcn_tanhf)
  return __builtin_amdgcn_tanhf(x);   // v_tanh_f32 (TRANS, co-executes with VALU)
#else
  return tanhf(x);
#endif
}

// Padded LDS strides (in floats): keep rows on distinct banks AND 16B-aligned
#define XS 264   // 264*4 B = 1056 B row stride (16B-aligned, 8-dword bank skew)
#define SS 516   // 516 dwords -> 4-bank skew per row for the WMMA A-loads

// ---------------------------------------------------------------------------
// Projection: Y = X * W^T + bias   (16 rows per workgroup, f32 WMMA)
// ---------------------------------------------------------------------------
__global__ __launch_bounds__(256)
void proj_kernel(const float* __restrict__ X, const float* __restrict__ W,
                 const float* __restrict__ bias, float* __restrict__ Y) {
  __shared__ __align__(16) float xs[16 * XS];
  const int tid  = threadIdx.x;
  const int lane = tid & 31;
  const int wave = tid >> 5;
  const long r0  = (long)blockIdx.x * 16;

  for (int idx = tid; idx < 16 * HH; idx += 256) {
    const int row = idx >> 8;
    const int col = idx & (HH - 1);
    xs[row * XS + col] = X[(r0 + row) * HH + col];
  }
  __syncthreads();

  const int arow = lane & 15;          // A: M = lane&15
  const int koff = (lane >> 4) << 1;   // A/B: K pair {0,1} or {2,3}
  const int n    = lane & 15;          // B/D: N = lane&15

  for (int half = 0; half < 2; ++half) {
    const int h0 = wave * 16 + half * 128;
    v8f acc = {};
    for (int k0 = 0; k0 < HH; k0 += 4) {
      v2f a, b;
      a.x = xs[arow * XS + k0 + koff];
      a.y = xs[arow * XS + k0 + koff + 1];
      b.x = W[(size_t)(h0 + n) * HH + k0 + koff];
      b.y = W[(size_t)(h0 + n) * HH + k0 + koff + 1];
      acc = WMMA_F32_4(a, b, acc);
    }
    const float bval = bias[h0 + n];
    float* yb = Y + (r0 + ((lane >> 4) << 3)) * HH + h0 + n;
#pragma unroll
    for (int j = 0; j < 8; ++j) yb[(size_t)j * HH] = acc[j] + bval;
  }
}

// ---------------------------------------------------------------------------
// Fused: scores = Wv . tanh(qh + kh) + bv  ->  mask -> softmax -> attn @ V
// One workgroup per (b, 16-q tile). kh row held in registers and reused
// across all 16 q rows; Wv register-resident; all hot loads are b128.
// ---------------------------------------------------------------------------
__global__ __launch_bounds__(256)
void attn_kernel(const float* __restrict__ qh, const float* __restrict__ kh,
                 const float* __restrict__ v,  const int* __restrict__ valid_lens,
                 const float* __restrict__ Wv, const float* __restrict__ bvp,
                 float* __restrict__ out) {
  __shared__ __align__(16) float qs[16 * XS];   // 16 x 256 qh tile
  __shared__ __align__(16) float sc[16 * SS];   // 16 x 512 scores / attn

  const int tid  = threadIdx.x;
  const int lane = tid & 31;
  const int wave = tid >> 5;
  const int b    = blockIdx.x >> 3;
  const int qt   = blockIdx.x & 7;
  const long q0  = (long)b * QQ + qt * 16;
  const float bv = bvp[0];
  const int  vl  = valid_lens[b];

  for (int idx = tid; idx < 16 * HH; idx += 256) {
    const int row = idx >> 8;
    const int col = idx & (HH - 1);
    qs[row * XS + col] = qh[(q0 + row) * HH + col];
  }
  __syncthreads();

  // ---- Phase 1: scores. Wv lives in 8 registers (h = c*128 + lane*4 + j) ---
  const float4* wv4 = (const float4*)Wv;
  const float4 wva = wv4[lane];
  const float4 wvb = wv4[32 + lane];
  const float* khb = kh + (size_t)b * KK * HH;

  for (int kk = wave; kk < KK; kk += 8) {       // wave owns k rows
    const float4* kr4 = (const float4*)(khb + (size_t)kk * HH);
    const float4 kra = kr4[lane];               // global_load_b128, coalesced
    const float4 krb = kr4[32 + lane];
#pragma unroll 4
    for (int qq = 0; qq < 16; ++qq) {           // reuse kh row across 16 q
      const float4 qa = *(const float4*)(qs + qq * XS + 4 * lane);
      const float4 qb = *(const float4*)(qs + qq * XS + 128 + 4 * lane);
      float partial =
          wva.x * fast_tanh(qa.x + kra.x) + wva.y * fast_tanh(qa.y + kra.y)
        + wva.z * fast_tanh(qa.z + kra.z) + wva.w * fast_tanh(qa.w + kra.w)
        + wvb.x * fast_tanh(qb.x + krb.x) + wvb.y * fast_tanh(qb.y + krb.y)
        + wvb.z * fast_tanh(qb.z + krb.z) + wvb.w * fast_tanh(qb.w + krb.w);
      for (int off = 16; off > 0; off >>= 1)
        partial += __shfl_xor(partial, off, 32);
      if (lane == 0) {
        float s = partial + bv;
        if (kk >= vl) s = NEGV;                 // sequence mask
        sc[qq * SS + kk] = s;
      }
    }
  }
  __syncthreads();

  // ---- Phase 2: softmax over K per q-row (wave w -> rows w and w+8) ----
  for (int rsel = 0; rsel < 2; ++rsel) {
    const int rq = wave + rsel * 8;
    float m = -3.4e38f;
    for (int c = lane; c < KK; c += 32) m = fmaxf(m, sc[rq * SS + c]);
    for (int off = 16; off > 0; off >>= 1) m = fmaxf(m, __shfl_xor(m, off, 32));
    float sum = 0.f;
    for (int c = lane; c < KK; c += 32) {
      const float e = __expf(sc[rq * SS + c] - m);   // v_exp_f32
      sc[rq * SS + c] = e;
      sum += e;
    }
    for (int off = 16; off > 0; off >>= 1) sum += __shfl_xor(sum, off, 32);
    const float inv = 1.f / sum;
    for (int c = lane; c < KK; c += 32) sc[rq * SS + c] *= inv;
  }
  __syncthreads();

  // ---- Phase 3: out(16x256) = attn(16x512) @ V(512x256) via f32 WMMA ----
  const float* vb  = v + (size_t)b * KK * DD;
  const int arow = lane & 15;
  const int koff = (lane >> 4) << 1;
  const int n    = lane & 15;
  for (int dt = 0; dt < 2; ++dt) {
    const int d0 = wave * 32 + dt * 16;
    v8f acc = {};
    for (int k0 = 0; k0 < KK; k0 += 4) {
      v2f a, bm;
      a.x  = sc[arow * SS + k0 + koff];
      a.y  = sc[arow * SS + k0 + koff + 1];
      bm.x = vb[(size_t)(k0 + koff) * DD + d0 + n];
      bm.y = vb[(size_t)(k0 + koff + 1) * DD + d0 + n];
      acc = WMMA_F32_4(a, bm, acc);
    }
    float* ob = out + (q0 + ((lane >> 4) << 3)) * DD + d0 + n;
#pragma unroll
    for (int j = 0; j < 8; ++j) ob[(size_t)j * DD] = acc[j];
  }
}

// ---------------------------------------------------------------------------
extern "C" void kernel_launch(void* const* d_in, const int* in_sizes, int n_in,
                              void* d_out, int out_size, void* d_ws, size_t ws_size,
                              hipStream_t stream) {
  (void)in_sizes; (void)n_in; (void)out_size; (void)ws_size;
  const float* q   = (const float*)d_in[0];
  const float* k   = (const float*)d_in[1];
  const float* v   = (const float*)d_in[2];
  const int*   vl  = (const int*)  d_in[3];
  const float* Wq  = (const float*)d_in[4];
  const float* bq  = (const float*)d_in[5];
  const float* Wk  = (const float*)d_in[6];
  const float* bk  = (const float*)d_in[7];
  const float* Wv  = (const float*)d_in[8];
  const float* bv  = (const float*)d_in[9];
  float* out = (float*)d_out;

  float* qh = (float*)d_ws;                        // B*Q*H = 2 MB
  float* kh = qh + (size_t)BB * QQ * HH;           // B*K*H = 8 MB

  proj_kernel<<<(BB * QQ) / 16, 256, 0, stream>>>(q, Wq, bq, qh);   // 128 WGs
  proj_kernel<<<(BB * KK) / 16, 256, 0, stream>>>(k, Wk, bk, kh);   // 512 WGs
  attn_kernel<<<BB * (QQ / 16), 256, 0, stream>>>(qh, kh, v, vl, Wv, bv, out);
}